// BatchCRF_4501125726840
// MI455X (gfx1250) — compile-verified
//
#include <hip/hip_runtime.h>

typedef float v2f __attribute__((ext_vector_type(2)));
typedef float v8f __attribute__((ext_vector_type(8)));

#define S_LEN 512
#define B_SZ  1024
#define T_TAG 50
#define TP    64    // padded tag dimension
#define LOG2E 1.4426950408889634f
#define LN2F  0.6931471805599453f

// ---------------------------------------------------------------------------
// Kernel 0: precompute E = exp(transitions) zero-padded to 64x64, and the
// gold path score per batch element.
// ---------------------------------------------------------------------------
__global__ __launch_bounds__(256) void crf_gold_kernel(
    const float* __restrict__ emissions,    // (S,B,T)
    const int*   __restrict__ tags,         // (S,B)
    const unsigned char* __restrict__ mask, // (S,B) bool bytes
    const float* __restrict__ transitions,  // (T,T)
    const float* __restrict__ start_trans,  // (T)
    const float* __restrict__ end_trans,    // (T)
    float* __restrict__ Epad,               // (64,64) out
    float* __restrict__ gold)               // (B) out
{
    const int tid = blockIdx.x * blockDim.x + threadIdx.x;  // 0..1023

    for (int i = tid; i < TP * TP; i += B_SZ) {
        int r = i >> 6, c = i & 63;
        Epad[i] = (r < T_TAG && c < T_TAG) ? expf(transitions[r * T_TAG + c]) : 0.0f;
    }

    const int b = tid;
    if (b >= B_SZ) return;

    int tg0 = tags[b];
    float g = start_trans[tg0] +
              __builtin_nontemporal_load(&emissions[(size_t)b * T_TAG + tg0]);
    int prev = tg0;
    int len = mask[b] ? 1 : 0;
#pragma unroll 2
    for (int t = 1; t < S_LEN; ++t) {
        int tg = tags[(size_t)t * B_SZ + b];
        float mt = mask[(size_t)t * B_SZ + b] ? 1.0f : 0.0f;
        float e  = __builtin_nontemporal_load(
                       &emissions[((size_t)t * B_SZ + b) * T_TAG + tg]);
        g += (transitions[prev * T_TAG + tg] + e) * mt;
        len += mask[(size_t)t * B_SZ + b] ? 1 : 0;
        prev = tg;
    }
    int last = tags[(size_t)(len - 1) * B_SZ + b];
    g += end_trans[last];
    gold[b] = g;
}

// ---------------------------------------------------------------------------
// Kernel 1: forward recursion, latency-optimized.
// Scores are held internally as u = (natural score) * log2(e) so exp2/log2
// native trans ops are used directly. The stabilizer m is refreshed every 4
// steps (exact for any m; drift-bound << exp2 range). One barrier per step.
// ---------------------------------------------------------------------------
__global__ __launch_bounds__(128) void crf_forward_kernel(
    const float* __restrict__ emissions,    // (S,B,T)
    const unsigned char* __restrict__ mask, // (S,B)
    const float* __restrict__ start_trans,  // (T)
    const float* __restrict__ end_trans,    // (T)
    const float* __restrict__ Epad,         // (64,64)
    const float* __restrict__ gold,         // (B)
    float* __restrict__ pertile)            // (B/16) out
{
    const int tile = blockIdx.x;        // batch tile 0..63
    const int wave = threadIdx.x >> 5;  // 0..3 -> N-tile
    const int lane = threadIdx.x & 31;
    const int hl   = lane >> 4;
    const int lcol = lane & 15;
    const int col  = wave * 16 + lcol;  // tag column j in [0,64)
    const int rowb = hl * 8;            // C-layout: row = v + 8*hl
    const int gb0  = tile * 16;

    const bool validc = (col < T_TAG);
    const int  colC   = validc ? col : (T_TAG - 1);

    __shared__ __align__(16) float pbuf[2][16][TP]; // P double buffer
    __shared__ __align__(16) float pmax[2][16][4];  // row-max partials
    __shared__ __align__(16) float psum[16][4];     // final partial sums

    // Resident B fragments of E (B layout: N = lane%16, K = 4G + 2*hl (+1))
    v2f bfrag[16];
#pragma unroll
    for (int g = 0; g < 16; ++g) {
        int k0 = 4 * g + 2 * hl;
        v2f bf;
        bf.x = Epad[k0 * TP + col];
        bf.y = Epad[(k0 + 1) * TP + col];
        bfrag[g] = bf;
    }

    // --- score init (base-2 scaled): u0 = (start + e0) * log2e ------------
    float st = start_trans[colC];
    v8f sc;
#pragma unroll
    for (int v = 0; v < 8; ++v) {
        float e = __builtin_nontemporal_load(
            &emissions[(size_t)(gb0 + rowb + v) * T_TAG + colC]);
        sc[v] = validc ? (st + e) * LOG2E : -1e30f;
    }

    // --- initial row-max exchange (uses pmax[0]) --------------------------
#pragma unroll
    for (int v = 0; v < 8; ++v) {
        float mv = sc[v];
        mv = fmaxf(mv, __shfl_xor(mv, 1, 32));
        mv = fmaxf(mv, __shfl_xor(mv, 2, 32));
        mv = fmaxf(mv, __shfl_xor(mv, 4, 32));
        mv = fmaxf(mv, __shfl_xor(mv, 8, 32));
        if (lcol == 0) pmax[0][rowb + v][wave] = mv;
    }
    __syncthreads();
    float mrow[8];
#pragma unroll
    for (int v = 0; v < 8; ++v) {
        float4 pm = *(const float4*)&pmax[0][rowb + v][0];
        mrow[v] = fmaxf(fmaxf(pm.x, pm.y), fmaxf(pm.z, pm.w));
    }

    // --- per-step pointers (constant 24-bit ioffsets inside the step) -----
    const float* ep = emissions + ((size_t)B_SZ + gb0 + rowb) * T_TAG + colC;
    const unsigned char* mp = mask + (size_t)B_SZ + gb0 + rowb;
    const char* pfp = (const char*)(emissions +
                      ((size_t)9 * B_SZ + gb0) * T_TAG) + lane * 128;

    int pb = 0;
#pragma unroll 1
    for (int t = 1; t < S_LEN; ++t) {
        // deep prefetch of emissions 8 steps ahead (25 x 128B lines/tile)
        if (wave == 3 && lane < 25 && t + 8 < S_LEN)
            __builtin_prefetch(pfp, 0, 1);
        pfp += (size_t)B_SZ * T_TAG * 4;

        // issue this step's emission + mask loads early
        float ev[8];
#pragma unroll
        for (int v = 0; v < 8; ++v)
            ev[v] = __builtin_nontemporal_load(&ep[v * T_TAG]);
        unsigned long long mk8 = *(const unsigned long long*)mp;

        const bool refresh = ((t & 3) == 0);

        // p = exp2(u - m_stale) -> LDS P tile
#pragma unroll
        for (int v = 0; v < 8; ++v)
            pbuf[pb][rowb + v][col] = __builtin_amdgcn_exp2f(sc[v] - mrow[v]);

        if (refresh) {
#pragma unroll
            for (int v = 0; v < 8; ++v) {
                float mv = sc[v];
                mv = fmaxf(mv, __shfl_xor(mv, 1, 32));
                mv = fmaxf(mv, __shfl_xor(mv, 2, 32));
                mv = fmaxf(mv, __shfl_xor(mv, 4, 32));
                mv = fmaxf(mv, __shfl_xor(mv, 8, 32));
                if (lcol == 0) pmax[pb][rowb + v][wave] = mv;
            }
        }
        __syncthreads();

        float nmrow[8];
        if (refresh) {
#pragma unroll
            for (int v = 0; v < 8; ++v) {
                float4 pm = *(const float4*)&pmax[pb][rowb + v][0];
                nmrow[v] = fmaxf(fmaxf(pm.x, pm.y), fmaxf(pm.z, pm.w));
            }
        }

        // D = P x E : 4 independent WMMA chains of depth 4
        v8f a0 = {}, a1 = {}, a2 = {}, a3 = {};
#pragma unroll
        for (int j = 0; j < 4; ++j) {
            v2f x0 = *(const v2f*)&pbuf[pb][lcol][16 * j + 0  + 2 * hl];
            v2f x1 = *(const v2f*)&pbuf[pb][lcol][16 * j + 4  + 2 * hl];
            v2f x2 = *(const v2f*)&pbuf[pb][lcol][16 * j + 8  + 2 * hl];
            v2f x3 = *(const v2f*)&pbuf[pb][lcol][16 * j + 12 + 2 * hl];
            a0 = __builtin_amdgcn_wmma_f32_16x16x4_f32(
                false, x0, false, bfrag[4 * j + 0], (short)0, a0, false, false);
            a1 = __builtin_amdgcn_wmma_f32_16x16x4_f32(
                false, x1, false, bfrag[4 * j + 1], (short)0, a1, false, false);
            a2 = __builtin_amdgcn_wmma_f32_16x16x4_f32(
                false, x2, false, bfrag[4 * j + 2], (short)0, a2, false, false);
            a3 = __builtin_amdgcn_wmma_f32_16x16x4_f32(
                false, x3, false, bfrag[4 * j + 3], (short)0, a3, false, false);
        }
        v8f acc = (a0 + a1) + (a2 + a3);

        // u' = e*log2e + m_stale + log2(acc), mask select
#pragma unroll
        for (int v = 0; v < 8; ++v) {
            float l2 = __builtin_amdgcn_logf(acc[v]);
            float nv = validc ? (fmaf(ev[v], LOG2E, mrow[v]) + l2) : -1e30f;
            bool mt = ((mk8 >> (8 * v)) & 0xffull) != 0ull;
            sc[v] = mt ? nv : sc[v];
        }
        if (refresh) {
#pragma unroll
            for (int v = 0; v < 8; ++v) mrow[v] = nmrow[v];
        }

        ep += (size_t)B_SZ * T_TAG;
        mp += B_SZ;
        pb ^= 1;
    }

    // --- final: logz = ln-space logsumexp_j(u/log2e + end_trans) ----------
    float et = validc ? end_trans[col] * LOG2E : 0.0f;
    float fv[8];
#pragma unroll
    for (int v = 0; v < 8; ++v) fv[v] = sc[v] + et;  // invalid cols stay -1e30

#pragma unroll
    for (int v = 0; v < 8; ++v) {
        float mv = fv[v];
        mv = fmaxf(mv, __shfl_xor(mv, 1, 32));
        mv = fmaxf(mv, __shfl_xor(mv, 2, 32));
        mv = fmaxf(mv, __shfl_xor(mv, 4, 32));
        mv = fmaxf(mv, __shfl_xor(mv, 8, 32));
        if (lcol == 0) pmax[0][rowb + v][wave] = mv;
    }
    __syncthreads();
#pragma unroll
    for (int v = 0; v < 8; ++v) {
        int r = rowb + v;
        float4 pm = *(const float4*)&pmax[0][r][0];
        float m = fmaxf(fmaxf(pm.x, pm.y), fmaxf(pm.z, pm.w));
        float e = __builtin_amdgcn_exp2f(fv[v] - m);
        e += __shfl_xor(e, 1, 32);
        e += __shfl_xor(e, 2, 32);
        e += __shfl_xor(e, 4, 32);
        e += __shfl_xor(e, 8, 32);
        if (lcol == 0) psum[r][wave] = e;
    }
    __syncthreads();

    if (wave == 0 && lane < 16) {
        int r = lane;
        float4 pm = *(const float4*)&pmax[0][r][0];
        float m = fmaxf(fmaxf(pm.x, pm.y), fmaxf(pm.z, pm.w));
        float4 ps = *(const float4*)&psum[r][0];
        float s = (ps.x + ps.y) + (ps.z + ps.w);
        float logz = (m + __builtin_amdgcn_logf(s)) * LN2F;  // back to nats
        float diff = logz - gold[gb0 + r];
        diff += __shfl_xor(diff, 1, 32);
        diff += __shfl_xor(diff, 2, 32);
        diff += __shfl_xor(diff, 4, 32);
        diff += __shfl_xor(diff, 8, 32);
        if (lane == 0) pertile[tile] = diff;
    }
}

// ---------------------------------------------------------------------------
// Kernel 2: deterministic fixed-order reduction of the 64 per-tile partials.
// ---------------------------------------------------------------------------
__global__ void crf_final_kernel(const float* __restrict__ pertile,
                                 float* __restrict__ out)
{
    if (threadIdx.x == 0 && blockIdx.x == 0) {
        float s = 0.0f;
        for (int i = 0; i < B_SZ / 16; ++i) s += pertile[i];
        out[0] = s * (1.0f / (float)B_SZ);
    }
}

// ---------------------------------------------------------------------------
extern "C" void kernel_launch(void* const* d_in, const int* in_sizes, int n_in,
                              void* d_out, int out_size, void* d_ws, size_t ws_size,
                              hipStream_t stream) {
    const float* emissions        = (const float*)d_in[0];
    const int*   tags             = (const int*)d_in[1];
    const unsigned char* mask     = (const unsigned char*)d_in[2];
    const float* transitions      = (const float*)d_in[3];
    const float* start_trans      = (const float*)d_in[4];
    const float* end_trans        = (const float*)d_in[5];

    float* ws      = (float*)d_ws;
    float* Epad    = ws;                    // 64*64 = 4096 floats
    float* gold    = ws + 4096;             // 1024 floats
    float* pertile = ws + 4096 + 1024;      // 64 floats

    crf_gold_kernel<<<4, 256, 0, stream>>>(emissions, tags, mask, transitions,
                                           start_trans, end_trans, Epad, gold);
    crf_forward_kernel<<<B_SZ / 16, 128, 0, stream>>>(emissions, mask,
                                                      start_trans, end_trans,
                                                      Epad, gold, pertile);
    crf_final_kernel<<<1, 32, 0, stream>>>(pertile, (float*)d_out);
}